// REARM_335007449938
// MI455X (gfx1250) — compile-verified
//
#include <hip/hip_runtime.h>
#include <math.h>

// ---------------------------------------------------------------------------
// Sizes from the reference
// ---------------------------------------------------------------------------
#define N_USERS 20000
#define N_ITEMS 30000
#define N_NODES (N_USERS + N_ITEMS)
#define DIM 64
#define KK 4
#define DV 2048
#define DT 384
#define NNZ_ADJ 1000000
#define NNZ_II  600000
#define NNZ_UU  400000

typedef __attribute__((ext_vector_type(2))) float v2f;
typedef __attribute__((ext_vector_type(8))) float v8f;

__device__ __forceinline__ float prelu_f(float x, float a) { return x >= 0.f ? x : a * x; }

// ---------------------------------------------------------------------------
// GEMM: C[M x N] = A[M x K] @ W[N x K]^T + bias[N], optional PReLU.
// A row stride = lda. C written at C[row*ldc + coff + n].
// V_WMMA_F32_16X16X4_F32; one wave owns a 16(M) x 64(N) strip; the block's
// shared 64xKC W-slab is staged in LDS (row pad +4 -> conflict-free ds_load_b64).
// M % 16 == 0, N % 64 == 0, K % 64 == 0 for all call sites here.
// ---------------------------------------------------------------------------
#define KC 64
#define BPAD 68   // 64 + 4 pad: float4-aligned rows, banks rotate by 4/row

__global__ void __launch_bounds__(256)
gemm_nt_wmma(const float* __restrict__ A, int lda,
             const float* __restrict__ W,
             const float* __restrict__ bias, float* __restrict__ C,
             int M, int K, int ldc, int coff,
             const float* __restrict__ alpha, int act)
{
  __shared__ float bs[64 * BPAD];
  const int tid    = threadIdx.x;
  const int lane   = tid & 31;
  const int wave   = tid >> 5;
  const int mtile  = blockIdx.x * 8 + wave;           // 16-row tile index
  const bool active = (mtile * 16 < M);               // wave-uniform
  const int nl     = lane & 15;
  const int hi     = lane >> 4;                       // 0: K={0,1}, 1: K={2,3}
  const int nbase  = blockIdx.y * 64;

  const int rowb = active ? (mtile * 16 + nl) : nl;
  const float* arow = A + (size_t)rowb * lda + 2 * hi;

  // cooperative W-tile staging: 4 threads per row, 16 floats each
  const int wrow = tid >> 2;                          // 0..63
  const int wseg = (tid & 3) * 16;                    // 0,16,32,48
  const float* wsrc = W + (size_t)(nbase + wrow) * K + wseg;
  float*       wdst = &bs[wrow * BPAD + wseg];

  v8f acc0 = {}, acc1 = {}, acc2 = {}, acc3 = {};

  for (int kc = 0; kc < K; kc += KC) {
    __syncthreads();
#pragma unroll
    for (int j = 0; j < 16; j += 4)
      *(float4*)(wdst + j) = *(const float4*)(wsrc + kc + j);
    __syncthreads();

    if (active) {
      if (kc + KC < K) __builtin_prefetch(arow + kc + KC, 0, 1);  // global_prefetch_b8
      const float* bbase = &bs[nl * BPAD + 2 * hi];
#pragma unroll 4
      for (int k0 = 0; k0 < KC; k0 += 4) {
        v2f a  = *(const v2f*)(arow + kc + k0);
        v2f b0 = *(const v2f*)(bbase + k0);
        v2f b1 = *(const v2f*)(bbase + 16 * BPAD + k0);
        v2f b2 = *(const v2f*)(bbase + 32 * BPAD + k0);
        v2f b3 = *(const v2f*)(bbase + 48 * BPAD + k0);
        acc0 = __builtin_amdgcn_wmma_f32_16x16x4_f32(false, a, false, b0, (short)0, acc0, false, false);
        acc1 = __builtin_amdgcn_wmma_f32_16x16x4_f32(false, a, false, b1, (short)0, acc1, false, false);
        acc2 = __builtin_amdgcn_wmma_f32_16x16x4_f32(false, a, false, b2, (short)0, acc2, false, false);
        acc3 = __builtin_amdgcn_wmma_f32_16x16x4_f32(false, a, false, b3, (short)0, acc3, false, false);
      }
    }
  }
  if (!active) return;

  const float av = act ? alpha[0] : 0.f;
  const int   mb = hi * 8;                            // D: lanes>=16 hold M=8..15
  v8f accs[4] = {acc0, acc1, acc2, acc3};
#pragma unroll
  for (int t = 0; t < 4; ++t) {
    const int   n  = nbase + t * 16 + nl;
    const float bv = bias ? bias[n] : 0.f;
#pragma unroll
    for (int r = 0; r < 8; ++r) {
      float v = accs[t][r] + bv;
      if (act) v = prelu_f(v, av);
      C[(size_t)(mtile * 16 + mb + r) * ldc + coff + n] = v;
    }
  }
}

// ---------------------------------------------------------------------------
// SpMM scatter: Y[row[e], :] += val[e] * X[col[e], :]   (one wave per edge)
// ---------------------------------------------------------------------------
__global__ void __launch_bounds__(256)
spmm_atomic(const int* __restrict__ rowi, const int* __restrict__ coli,
            const float* __restrict__ val, const float* __restrict__ X,
            float* __restrict__ Y, int nnz, int F)
{
  const int e    = (int)((blockIdx.x * (size_t)blockDim.x + threadIdx.x) >> 5);
  const int lane = threadIdx.x & 31;
  if (e >= nnz) return;
  const int   r = rowi[e], c = coli[e];
  const float v = val[e];
  const float* xr = X + (size_t)c * F;
  float*       yr = Y + (size_t)r * F;
  for (int f = lane; f < F; f += 32) atomicAdd(&yr[f], v * xr[f]);
}

// ---------------------------------------------------------------------------
// Row-wise L2 normalize in place (wave per row)
// ---------------------------------------------------------------------------
__global__ void __launch_bounds__(256)
l2norm_rows(float* __restrict__ X, int n, int F)
{
  const int r    = (int)((blockIdx.x * (size_t)blockDim.x + threadIdx.x) >> 5);
  const int lane = threadIdx.x & 31;
  if (r >= n) return;
  float* xr = X + (size_t)r * F;
  float  s  = 0.f;
  for (int f = lane; f < F; f += 32) { float t = xr[f]; s += t * t; }
  for (int m = 16; m; m >>= 1) s += __shfl_xor(s, m, 32);
  const float inv = 1.f / fmaxf(sqrtf(s), 1e-12f);
  for (int f = lane; f < F; f += 32) xr[f] *= inv;
}

// ---------------------------------------------------------------------------
// Tiny attention + residual + LayerNorm + PReLU. Wave per row (D=64).
// residual == XKV for all four uses in the reference.
// ---------------------------------------------------------------------------
__global__ void __launch_bounds__(256)
attn_ln_prelu(const float* __restrict__ XQ, int ldq,
              const float* __restrict__ XKV, int ldkv,
              const float* __restrict__ aw, const float* __restrict__ ab,
              const float* __restrict__ aow, const float* __restrict__ aob, int idx,
              const float* __restrict__ gamma, const float* __restrict__ beta,
              const float* __restrict__ alpha,
              float* __restrict__ OUT, int n)
{
  const int r    = (int)((blockIdx.x * (size_t)blockDim.x + threadIdx.x) >> 5);
  const int lane = threadIdx.x & 31;
  if (r >= n) return;
  const float w0 = aw[idx * 3 + 0], w1 = aw[idx * 3 + 1], w2 = aw[idx * 3 + 2];
  const float b0 = ab[idx * 3 + 0], b1 = ab[idx * 3 + 1], b2 = ab[idx * 3 + 2];
  const float wo = aow[idx], bo = aob[idx];

  const float* xq  = XQ  + (size_t)r * ldq;
  const float* xkv = XKV + (size_t)r * ldkv;
  const float q0  = xq[lane] * w0 + b0, q1 = xq[lane + 32] * w0 + b0;
  const float kv0 = xkv[lane],          kv1 = xkv[lane + 32];
  const float k0  = kv0 * w1 + b1,      k1  = kv1 * w1 + b1;
  const float v0  = kv0 * w2 + b2,      v1  = kv1 * w2 + b2;

  float kmax = fmaxf(k0, k1), kmin = fminf(k0, k1);
  for (int m = 16; m; m >>= 1) {
    kmax = fmaxf(kmax, __shfl_xor(kmax, m, 32));
    kmin = fminf(kmin, __shfl_xor(kmin, m, 32));
  }

  float o[2];
  const float qv[2] = {q0, q1};
#pragma unroll
  for (int j = 0; j < 2; ++j) {
    const float q  = qv[j];
    const float mx = (q >= 0.f) ? q * kmax : q * kmin;   // exact max of q*k_m
    float se = 0.f, sv = 0.f;
    for (int m = 0; m < 32; ++m) {
      float km = __shfl(k0, m, 32), vm = __shfl(v0, m, 32);
      float e  = __expf(q * km - mx); se += e; sv += e * vm;
      km = __shfl(k1, m, 32); vm = __shfl(v1, m, 32);
      e  = __expf(q * km - mx); se += e; sv += e * vm;
    }
    o[j] = (sv / se) * wo + bo;
  }

  float y0 = kv0 + o[0], y1 = kv1 + o[1];
  float s = y0 + y1;
  for (int m = 16; m; m >>= 1) s += __shfl_xor(s, m, 32);
  const float mu = s * (1.f / 64.f);
  const float d0 = y0 - mu, d1 = y1 - mu;
  float vs = d0 * d0 + d1 * d1;
  for (int m = 16; m; m >>= 1) vs += __shfl_xor(vs, m, 32);
  const float inv = rsqrtf(vs * (1.f / 64.f) + 1e-5f);
  const float a   = alpha[0];
  float z0 = prelu_f(d0 * inv * gamma[lane] + beta[lane], a);
  float z1 = prelu_f(d1 * inv * gamma[lane + 32] + beta[lane + 32], a);
  OUT[(size_t)r * 64 + lane]      = z0;
  OUT[(size_t)r * 64 + lane + 32] = z1;
}

// ---------------------------------------------------------------------------
// Strided copy with optional PReLU: Y[r*ldy+f] = act(X[r*ldx+f])
// ---------------------------------------------------------------------------
__global__ void __launch_bounds__(256)
copy_act(const float* __restrict__ X, int ldx, float* __restrict__ Y, int ldy,
         int n, int F, const float* __restrict__ alpha, int act)
{
  const long i = blockIdx.x * (long)blockDim.x + threadIdx.x;
  const long total = (long)n * F;
  if (i >= total) return;
  const int r = (int)(i / F), f = (int)(i % F);
  float v = X[(size_t)r * ldx + f];
  if (act) v = prelu_f(v, alpha[0]);
  Y[(size_t)r * ldy + f] = v;
}

// o = (a + b + c) / 3
__global__ void __launch_bounds__(256)
combine3(const float* __restrict__ a, const float* __restrict__ b,
         const float* __restrict__ c, float* __restrict__ o, long count)
{
  const long i = blockIdx.x * (long)blockDim.x + threadIdx.x;
  if (i >= count) return;
  o[i] = (a[i] + b[i] + c[i]) * (1.f / 3.f);
}

// ---------------------------------------------------------------------------
// Per-node bottleneck: share[n,d] = sum_k ( sum_d' id[n,d'] m1[n,d',k] ) m2[n,k,d]
// m1: 256 = [d*4+k], m2: 256 = [k*64+d]. Wave per node. ID has row stride ldid.
// ---------------------------------------------------------------------------
__global__ void __launch_bounds__(256)
meta_share(const float* __restrict__ ID, int ldid,
           const float* __restrict__ M1, const float* __restrict__ M2,
           float* __restrict__ OUT, int n)
{
  const int r    = (int)((blockIdx.x * (size_t)blockDim.x + threadIdx.x) >> 5);
  const int lane = threadIdx.x & 31;
  if (r >= n) return;
  const float* id = ID + (size_t)r * ldid;
  const float* m1 = M1 + (size_t)r * 256;
  const float* m2 = M2 + (size_t)r * 256;
  const float id0 = id[lane], id1 = id[lane + 32];
  float t[KK];
#pragma unroll
  for (int k = 0; k < KK; ++k) {
    float s = id0 * m1[lane * 4 + k] + id1 * m1[(lane + 32) * 4 + k];
    for (int m = 16; m; m >>= 1) s += __shfl_xor(s, m, 32);
    t[k] = s;
  }
  float o0 = 0.f, o1 = 0.f;
#pragma unroll
  for (int k = 0; k < KK; ++k) {
    o0 += t[k] * m2[k * 64 + lane];
    o1 += t[k] * m2[k * 64 + lane + 32];
  }
  OUT[(size_t)r * 64 + lane]      = o0;
  OUT[(size_t)r * 64 + lane + 32] = o1;
}

// fin_all rows: [feat_v(64) | feat_t(64) | id(64)] (ld 192)
// out[r,0:64]=prelu(feat_v)+id ; [64:128]=prelu(feat_t)+id ; [128:192]=prelu(share)+id
__global__ void __launch_bounds__(256)
final_assemble(const float* __restrict__ fin_all, const float* __restrict__ share,
               const float* __restrict__ alpha, float* __restrict__ out, int n)
{
  const long i = blockIdx.x * (long)blockDim.x + threadIdx.x;
  const long total = (long)n * 64;
  if (i >= total) return;
  const int r = (int)(i >> 6), f = (int)(i & 63);
  const float a   = alpha[0];
  const float idv = fin_all[(size_t)r * 192 + 128 + f];
  float v = prelu_f(fin_all[(size_t)r * 192 + f], a);
  float t = prelu_f(fin_all[(size_t)r * 192 + 64 + f], a);
  float s = prelu_f(share[i], a);
  out[(size_t)r * 192 + f]       = v + idv;
  out[(size_t)r * 192 + 64 + f]  = t + idv;
  out[(size_t)r * 192 + 128 + f] = s + idv;
}

// ---------------------------------------------------------------------------
// Host-side orchestration
// ---------------------------------------------------------------------------
static inline void gemm(const float* A, int lda, const float* W, const float* b, float* C,
                        int M, int N, int K, int ldc, int coff,
                        const float* alpha, int act, hipStream_t s)
{
  dim3 grid((M / 16 + 7) / 8, N / 64);
  gemm_nt_wmma<<<grid, 256, 0, s>>>(A, lda, W, b, C, M, K, ldc, coff, alpha, act);
}

static inline void spmm(const int* r, const int* c, const float* v,
                        const float* X, float* Y, int nnz, int F, hipStream_t s)
{
  spmm_atomic<<<(nnz + 7) / 8, 256, 0, s>>>(r, c, v, X, Y, nnz, F);
}

extern "C" void kernel_launch(void* const* d_in, const int* in_sizes, int n_in,
                              void* d_out, int out_size, void* d_ws, size_t ws_size,
                              hipStream_t stream)
{
  const float* user_id_emb  = (const float*)d_in[0];
  const float* item_id_emb  = (const float*)d_in[1];
  const float* image_feat   = (const float*)d_in[2];
  const float* text_feat    = (const float*)d_in[3];
  const float* user_v_pref  = (const float*)d_in[4];
  const float* user_t_pref  = (const float*)d_in[5];
  const float* img_i_w = (const float*)d_in[6];
  const float* img_i_b = (const float*)d_in[7];
  const float* txt_i_w = (const float*)d_in[8];
  const float* txt_i_b = (const float*)d_in[9];
  const float* img_u_w = (const float*)d_in[10];
  const float* img_u_b = (const float*)d_in[11];
  const float* txt_u_w = (const float*)d_in[12];
  const float* txt_u_b = (const float*)d_in[13];
  const float* ln_gamma  = (const float*)d_in[14];
  const float* ln_beta   = (const float*)d_in[15];
  const float* prl_alpha = (const float*)d_in[16];
  const float* attn_in_w  = (const float*)d_in[17];
  const float* attn_in_b  = (const float*)d_in[18];
  const float* attn_out_w = (const float*)d_in[19];
  const float* attn_out_b = (const float*)d_in[20];
  const float* mlp_pre_w = (const float*)d_in[21];
  const float* mlp_pre_b = (const float*)d_in[22];
  const float* mlp_out_w = (const float*)d_in[23];
  const float* mlp_out_b = (const float*)d_in[24];
  const float* mlp_alpha = (const float*)d_in[25];
  const float* meta_u_w = (const float*)d_in[26];
  const float* meta_u_b = (const float*)d_in[27];
  const float* meta_i_w = (const float*)d_in[28];
  const float* meta_i_b = (const float*)d_in[29];
  const float* adj_val = (const float*)d_in[30];
  const float* ii_val  = (const float*)d_in[31];
  const float* uu_val  = (const float*)d_in[32];
  const int* adj_row = (const int*)d_in[33];
  const int* adj_col = (const int*)d_in[34];
  const int* ii_row  = (const int*)d_in[35];
  const int* ii_col  = (const int*)d_in[36];
  const int* uu_row  = (const int*)d_in[37];
  const int* uu_col  = (const int*)d_in[38];
  float* out = (float*)d_out;

  // ---- workspace layout (floats) -----------------------------------------
  float* ws = (float*)d_ws;
  size_t off = 0;
  auto alloc = [&](size_t n) { float* p = ws + off; off += n; return p; };
  // Region A (dead by MLP stage; h/m1/m2 alias into it)
  float* item_cat  = alloc((size_t)N_ITEMS * 192);
  float* item_prop = alloc((size_t)N_ITEMS * 192);
  float* user_cat  = alloc((size_t)N_USERS * 192);
  float* user_prop = alloc((size_t)N_USERS * 192);
  float* item_v    = alloc((size_t)N_ITEMS * 64);
  float* item_t    = alloc((size_t)N_ITEMS * 64);
  float* item_t2v  = alloc((size_t)N_ITEMS * 64);
  float* item_v2t  = alloc((size_t)N_ITEMS * 64);
  // Region B: fused [feat(128) | id(64)] node state, 192-wide
  float* ego_all = alloc((size_t)N_NODES * 192);
  float* p1      = alloc((size_t)N_NODES * 192);
  float* p2      = alloc((size_t)N_NODES * 192);
  float* fin_all = alloc((size_t)N_NODES * 192);
  // Region D
  float* tu    = alloc((size_t)N_USERS * 64);
  float* ti    = alloc((size_t)N_ITEMS * 64);
  float* share = alloc((size_t)N_NODES * 64);
  // Aliases into Region A (safe: region A fully dead once ego_all is built)
  float* h_buf  = ws;                       // 30000*256 = 7.68M floats
  float* m1_buf = ws + (size_t)7680000;     // 7.68M floats
  float* m2_buf = ws + (size_t)15360000;    // 7.68M floats

  auto nblk = [](long total) { return (unsigned)((total + 255) / 256); };

  // ---- 1) projections + concat -------------------------------------------
  copy_act<<<nblk((long)N_ITEMS * 64), 256, 0, stream>>>(item_id_emb, 64, item_cat, 192, N_ITEMS, 64, nullptr, 0);
  gemm(image_feat, DV, img_i_w, img_i_b, item_cat, N_ITEMS, 64, DV, 192, 64, nullptr, 0, stream);
  gemm(text_feat,  DT, txt_i_w, txt_i_b, item_cat, N_ITEMS, 64, DT, 192, 128, nullptr, 0, stream);
  copy_act<<<nblk((long)N_USERS * 64), 256, 0, stream>>>(user_id_emb, 64, user_cat, 192, N_USERS, 64, nullptr, 0);
  gemm(user_v_pref, DV, img_u_w, img_u_b, user_cat, N_USERS, 64, DV, 192, 64, nullptr, 0, stream);
  gemm(user_t_pref, DT, txt_u_w, txt_u_b, user_cat, N_USERS, 64, DT, 192, 128, nullptr, 0, stream);

  // ---- 2) item-item / user-user propagation + l2norm ---------------------
  hipMemsetAsync(item_prop, 0, (size_t)N_ITEMS * 192 * sizeof(float), stream);
  spmm(ii_row, ii_col, ii_val, item_cat, item_prop, NNZ_II, 192, stream);
  l2norm_rows<<<(N_ITEMS + 7) / 8, 256, 0, stream>>>(item_prop, N_ITEMS, 192);
  hipMemsetAsync(user_prop, 0, (size_t)N_USERS * 192 * sizeof(float), stream);
  spmm(uu_row, uu_col, uu_val, user_cat, user_prop, NNZ_UU, 192, stream);
  l2norm_rows<<<(N_USERS + 7) / 8, 256, 0, stream>>>(user_prop, N_USERS, 192);

  // ---- 3) attention blocks (gnn_iv = cols 64:128, gnn_it = cols 128:192) --
  const unsigned ablk = (N_ITEMS + 7) / 8;
  attn_ln_prelu<<<ablk, 256, 0, stream>>>(item_prop + 64, 192, item_prop + 64, 192,
      attn_in_w, attn_in_b, attn_out_w, attn_out_b, 0, ln_gamma, ln_beta, prl_alpha, item_v, N_ITEMS);
  attn_ln_prelu<<<ablk, 256, 0, stream>>>(item_prop + 128, 192, item_prop + 128, 192,
      attn_in_w, attn_in_b, attn_out_w, attn_out_b, 1, ln_gamma, ln_beta, prl_alpha, item_t, N_ITEMS);
  attn_ln_prelu<<<ablk, 256, 0, stream>>>(item_t, 64, item_v, 64,
      attn_in_w, attn_in_b, attn_out_w, attn_out_b, 2, ln_gamma, ln_beta, prl_alpha, item_t2v, N_ITEMS);
  attn_ln_prelu<<<ablk, 256, 0, stream>>>(item_v, 64, item_t, 64,
      attn_in_w, attn_in_b, attn_out_w, attn_out_b, 3, ln_gamma, ln_beta, prl_alpha, item_v2t, N_ITEMS);

  // ---- 4) build fused ego state [feat_v | feat_t | id], 192-wide ----------
  copy_act<<<nblk((long)N_USERS * 64), 256, 0, stream>>>(user_prop + 64, 192, ego_all, 192, N_USERS, 64, prl_alpha, 1);
  copy_act<<<nblk((long)N_USERS * 64), 256, 0, stream>>>(user_prop + 128, 192, ego_all + 64, 192, N_USERS, 64, prl_alpha, 1);
  copy_act<<<nblk((long)N_USERS * 64), 256, 0, stream>>>(user_prop, 192, ego_all + 128, 192, N_USERS, 64, nullptr, 0);
  copy_act<<<nblk((long)N_ITEMS * 64), 256, 0, stream>>>(item_t2v, 64, ego_all + (size_t)N_USERS * 192, 192, N_ITEMS, 64, nullptr, 0);
  copy_act<<<nblk((long)N_ITEMS * 64), 256, 0, stream>>>(item_v2t, 64, ego_all + (size_t)N_USERS * 192 + 64, 192, N_ITEMS, 64, nullptr, 0);
  copy_act<<<nblk((long)N_ITEMS * 64), 256, 0, stream>>>(item_prop, 192, ego_all + (size_t)N_USERS * 192 + 128, 192, N_ITEMS, 64, nullptr, 0);

  // ---- 5) 2-layer propagation on adj (feat+id fused), mean of [x0,x1,x2] --
  hipMemsetAsync(p1, 0, (size_t)N_NODES * 192 * sizeof(float), stream);
  spmm(adj_row, adj_col, adj_val, ego_all, p1, NNZ_ADJ, 192, stream);
  hipMemsetAsync(p2, 0, (size_t)N_NODES * 192 * sizeof(float), stream);
  spmm(adj_row, adj_col, adj_val, p1, p2, NNZ_ADJ, 192, stream);
  combine3<<<nblk((long)N_NODES * 192), 256, 0, stream>>>(ego_all, p1, p2, fin_all, (long)N_NODES * 192);

  // ---- 6) meta linears (A = feat part of fin_all, K=128, lda=192) ---------
  gemm(fin_all, 192, meta_u_w, meta_u_b, tu, N_USERS, 64, 128, 64, 0, nullptr, 0, stream);
  gemm(fin_all + (size_t)N_USERS * 192, 192, meta_i_w, meta_i_b, ti, N_ITEMS, 64, 128, 64, 0, nullptr, 0, stream);

  // ---- 7) MLPs + per-node bottleneck einsums ------------------------------
  gemm(tu, 64, mlp_pre_w + 0 * 256 * 64, mlp_pre_b + 0 * 256, h_buf, N_USERS, 256, 64, 256, 0, mlp_alpha + 0, 1, stream);
  gemm(h_buf, 256, mlp_out_w + 0 * 256 * 256, mlp_out_b + 0 * 256, m1_buf, N_USERS, 256, 256, 256, 0, nullptr, 0, stream);
  l2norm_rows<<<(N_USERS + 7) / 8, 256, 0, stream>>>(m1_buf, N_USERS, 256);
  gemm(tu, 64, mlp_pre_w + 1 * 256 * 64, mlp_pre_b + 1 * 256, h_buf, N_USERS, 256, 64, 256, 0, mlp_alpha + 1, 1, stream);
  gemm(h_buf, 256, mlp_out_w + 1 * 256 * 256, mlp_out_b + 1 * 256, m2_buf, N_USERS, 256, 256, 256, 0, nullptr, 0, stream);
  l2norm_rows<<<(N_USERS + 7) / 8, 256, 0, stream>>>(m2_buf, N_USERS, 256);
  meta_share<<<(N_USERS + 7) / 8, 256, 0, stream>>>(fin_all + 128, 192, m1_buf, m2_buf, share, N_USERS);

  gemm(ti, 64, mlp_pre_w + 2 * 256 * 64, mlp_pre_b + 2 * 256, h_buf, N_ITEMS, 256, 64, 256, 0, mlp_alpha + 2, 1, stream);
  gemm(h_buf, 256, mlp_out_w + 2 * 256 * 256, mlp_out_b + 2 * 256, m1_buf, N_ITEMS, 256, 256, 256, 0, nullptr, 0, stream);
  l2norm_rows<<<(N_ITEMS + 7) / 8, 256, 0, stream>>>(m1_buf, N_ITEMS, 256);
  gemm(ti, 64, mlp_pre_w + 3 * 256 * 64, mlp_pre_b + 3 * 256, h_buf, N_ITEMS, 256, 64, 256, 0, mlp_alpha + 3, 1, stream);
  gemm(h_buf, 256, mlp_out_w + 3 * 256 * 256, mlp_out_b + 3 * 256, m2_buf, N_ITEMS, 256, 256, 256, 0, nullptr, 0, stream);
  l2norm_rows<<<(N_ITEMS + 7) / 8, 256, 0, stream>>>(m2_buf, N_ITEMS, 256);
  meta_share<<<(N_ITEMS + 7) / 8, 256, 0, stream>>>(fin_all + (size_t)N_USERS * 192 + 128, 192, m1_buf, m2_buf,
                                                    share + (size_t)N_USERS * 64, N_ITEMS);

  // ---- 8) final assembly --------------------------------------------------
  final_assemble<<<nblk((long)N_NODES * 64), 256, 0, stream>>>(fin_all, share, prl_alpha, out, N_NODES);
}